// YoloLayer_38809324487147
// MI455X (gfx1250) — compile-verified
//
#include <hip/hip_runtime.h>

#define NB      128
#define NA      3
#define NC      80
#define NH      19
#define NW      19
#define MAX_GT  50
#define PLANE   (NH * NW)          // 361
#define CELLS_B (NA * PLANE)       // 1083
#define NTOT    (NB * CELLS_B)     // 138624
#define REC     12                 // floats per GT record in workspace
#define RECTOT  (MAX_GT * REC)     // 600 floats = 2400 B per batch

typedef float v2f __attribute__((ext_vector_type(2)));
typedef float v8f __attribute__((ext_vector_type(8)));

__device__ __forceinline__ float sigf(float x) {
    return 1.0f / (1.0f + __expf(-x));
}

// ---------------------------------------------------------------------------
// Kernel 1: per-batch ground-truth preprocessing (tiny: 128 threads).
// Record layout (floats unless noted):
//  [0..3] gx,gy,gw,gh  [4..7] tc0..tc3  [8] flat(int) [9] tcls(int)
//  [10] valid (0/1)    [11] pad
// ---------------------------------------------------------------------------
__global__ void yolo_prep(const float* __restrict__ target,
                          const float* __restrict__ anchors,
                          float* __restrict__ ws,
                          float* __restrict__ out) {
    int b = blockIdx.x * blockDim.x + threadIdx.x;
    if (b == 0) out[0] = 0.0f;           // zero the loss accumulator each call
    if (b >= NB) return;

    const float* tb = target + b * MAX_GT * 5;
    float*       rc = ws + b * RECTOT;
    int*         ri = (int*)rc;

    float aw[3], ah[3];
#pragma unroll
    for (int n = 0; n < 3; ++n) { aw[n] = anchors[2 * n]; ah[n] = anchors[2 * n + 1]; }

    int validAll = 1;
    for (int t = 0; t < MAX_GT; ++t) {
        float cls = tb[t * 5 + 0];
        float x   = tb[t * 5 + 1];
        float y   = tb[t * 5 + 2];
        float w   = tb[t * 5 + 3];
        float h   = tb[t * 5 + 4];
        validAll = validAll && (x != 0.0f);      // cumprod(x != 0)

        float gx = x * (float)NW, gy = y * (float)NH;
        float gw = w * (float)NW, gh = h * (float)NH;   // NET_W/STRIDE == NW

        // argmax IoU of (0,0,gw,gh) vs (0,0,aw,ah); first max wins on ties
        float best = -1.0f; int bn = 0;
#pragma unroll
        for (int n = 0; n < 3; ++n) {
            float inter = fminf(gw, aw[n]) * fminf(gh, ah[n]);
            float uni   = gw * gh + aw[n] * ah[n] - inter;
            float iou   = inter / fmaxf(uni, 1e-12f);
            if (iou > best) { best = iou; bn = n; }
        }

        int gi = (int)floorf(gx), gj = (int)floorf(gy);
        int flat = ((b * NA + bn) * NH + gj) * NW + gi;
        if (!validAll) flat = NTOT;              // out-of-range sentinel (drop)

        float sgw = validAll ? gw : 1.0f;
        float sgh = validAll ? gh : 1.0f;

        rc[t * REC + 0]  = gx;
        rc[t * REC + 1]  = gy;
        rc[t * REC + 2]  = gw;
        rc[t * REC + 3]  = gh;
        rc[t * REC + 4]  = gx - floorf(gx);
        rc[t * REC + 5]  = gy - floorf(gy);
        rc[t * REC + 6]  = __logf(sgw / aw[bn]);
        rc[t * REC + 7]  = __logf(sgh / ah[bn]);
        ri[t * REC + 8]  = flat;
        ri[t * REC + 9]  = (int)cls;
        rc[t * REC + 10] = validAll ? 1.0f : 0.0f;
        rc[t * REC + 11] = 0.0f;
    }
}

// ---------------------------------------------------------------------------
// Kernel 2: main loss kernel. One thread per (anchor, y, x) cell of one batch.
// grid = (ceil(1083/256), 128), block = 256 (8 wave32s).
// GT table is staged global->LDS with GLOBAL_LOAD_ASYNC_TO_LDS_B128 so the
// staging overlaps the strided box-channel loads; drained via ASYNCcnt.
// ---------------------------------------------------------------------------
__global__ __launch_bounds__(256) void yolo_main(const float* __restrict__ output,
                                                 const float* __restrict__ ws,
                                                 const float* __restrict__ anchors,
                                                 float* __restrict__ out) {
    __shared__ __align__(16) float s[RECTOT];    // 2400 B GT table for this batch

    const int b = blockIdx.y;

    // ---- async stage: 150 lanes move 16 B each (600 floats total) ----------
    {
        const int k = threadIdx.x;
        if (k < RECTOT / 4) {
            unsigned int       lds_off = (unsigned int)(size_t)(&s[0]) + 16u * k;
            unsigned long long gaddr   =
                (unsigned long long)(const void*)(ws + (size_t)b * RECTOT + 4u * k);
            asm volatile("global_load_async_to_lds_b128 %0, %1, off"
                         :: "v"(lds_off), "v"(gaddr)
                         : "memory");
        }
    }

    // ---- issue the latency-critical box-channel loads while staging runs ---
    const int c   = blockIdx.x * blockDim.x + threadIdx.x;   // cell within batch
    const float act = (c < CELLS_B) ? 1.0f : 0.0f;
    const int cc  = (c < CELLS_B) ? c : (CELLS_B - 1);       // clamp for safe loads

    const int a   = cc / PLANE;
    const int rem = cc - a * PLANE;
    const int j   = rem / NW;
    const int i   = rem - j * NW;

    const float aw = anchors[2 * a], ah = anchors[2 * a + 1];

    const float* base = output + ((size_t)b * (NA * (5 + NC)) + (size_t)a * (5 + NC)) * PLANE + rem;
    const float txr = base[0 * PLANE];
    const float tyr = base[1 * PLANE];
    const float tw  = base[2 * PLANE];
    const float th  = base[3 * PLANE];
    const float cfr = base[4 * PLANE];

    const float sx   = sigf(txr);
    const float sy   = sigf(tyr);
    const float conf = sigf(cfr);
    const float pw   = __expf(tw) * aw;
    const float ph   = __expf(th) * ah;
    const float px   = sx + (float)i;
    const float py   = sy + (float)j;

    const float ax1 = px - 0.5f * pw, ax2 = px + 0.5f * pw;
    const float ay1 = py - 0.5f * ph, ay2 = py + 0.5f * ph;
    const float pArea = pw * ph;

    // ---- drain async LDS fill, then barrier so every wave sees the table ---
    asm volatile("s_wait_asynccnt 0" ::: "memory");
    __syncthreads();

    const int myflat = b * CELLS_B + cc;

    float maxiou = 0.0f;
    float coordm = 0.0f, tconf = 0.0f;
    float tc0 = 0.0f, tc1 = 0.0f, tc2 = 0.0f, tc3 = 0.0f;
    int   tcls = 0;

    for (int t = 0; t < MAX_GT; ++t) {
        const float v  = s[t * REC + 10];
        const float gx = s[t * REC + 0], gy = s[t * REC + 1];
        const float gw = s[t * REC + 2], gh = s[t * REC + 3];
        const float bx1 = gx - 0.5f * gw, bx2 = gx + 0.5f * gw;
        const float by1 = gy - 0.5f * gh, by2 = gy + 0.5f * gh;
        const float iw = fmaxf(fminf(ax2, bx2) - fmaxf(ax1, bx1), 0.0f);
        const float ih = fmaxf(fminf(ay2, by2) - fmaxf(ay1, by1), 0.0f);
        const float inter = iw * ih;
        const float uni   = pArea + gw * gh - inter;
        const float iou   = inter / fmaxf(uni, 1e-12f);
        maxiou = fmaxf(maxiou, v * iou);

        const int flat = ((const int*)s)[t * REC + 8];
        if (flat == myflat) {                    // reverse scatter (last match)
            coordm = 1.0f; tconf = 1.0f;
            tc0 = s[t * REC + 4]; tc1 = s[t * REC + 5];
            tc2 = s[t * REC + 6]; tc3 = s[t * REC + 7];
            tcls = ((const int*)s)[t * REC + 9];
        }
    }

    coordm *= act;
    float cmask = (maxiou > 0.5f) ? 0.0f : 1.0f;
    if (coordm > 0.0f) cmask = 1.0f;             // scatter-set overrides

    float loss = 0.0f;
    // coord L2 (masked)
    {
        const float d0 = sx - tc0, d1 = sy - tc1, d2 = tw - tc2, d3 = th - tc3;
        loss += 0.5f * coordm * (d0 * d0 + d1 * d1 + d2 * d2 + d3 * d3);
    }
    // conf L2
    {
        const float dc = conf - (coordm > 0.0f ? tconf : 0.0f);
        loss += cmask * dc * dc;
    }
    // class CE only for matched cells (<= 6400 of 138624): 13x traffic saving
    if (coordm > 0.0f) {
        const float* cl = base + 5 * PLANE;
        float m = -3.402823466e38f;
#pragma unroll 4
        for (int k = 0; k < NC; ++k) m = fmaxf(m, cl[k * PLANE]);
        float ssum = 0.0f;
#pragma unroll 4
        for (int k = 0; k < NC; ++k) ssum += __expf(cl[k * PLANE] - m);
        const float lt = cl[tcls * PLANE];
        loss += -(lt - m - __logf(ssum));
    }

    // ------------------------------------------------------------------
    // Wave reduction through the matrix pipe: V_WMMA_F32_16X16X4_F32 with
    // A = lane partials (16x4 layout) and B = ones => D[m][n] = rowsum(m).
    // Lane 0 holds D[0..7][0], lane 16 holds D[8..15][0]; their sums add
    // to the wave total. All 256 threads reach this point (EXEC all ones).
    // ------------------------------------------------------------------
    const float partial = act * loss;
    v2f A; A[0] = partial; A[1] = 0.0f;
    v2f B; B[0] = 1.0f;    B[1] = 1.0f;
    v8f C = {};
    C = __builtin_amdgcn_wmma_f32_16x16x4_f32(false, A, false, B,
                                              (short)0, C, false, false);
    const float s8 = C[0] + C[1] + C[2] + C[3] + C[4] + C[5] + C[6] + C[7];
    const int lane = threadIdx.x & 31;
    if ((lane & 15) == 0) atomicAdd(out, s8);
}

// ---------------------------------------------------------------------------
extern "C" void kernel_launch(void* const* d_in, const int* in_sizes, int n_in,
                              void* d_out, int out_size, void* d_ws, size_t ws_size,
                              hipStream_t stream) {
    const float* output  = (const float*)d_in[0];
    const float* target  = (const float*)d_in[1];
    const float* anchors = (const float*)d_in[2];
    float*       out     = (float*)d_out;
    float*       ws      = (float*)d_ws;        // needs NB*MAX_GT*REC*4 = 300 KB

    yolo_prep<<<1, 128, 0, stream>>>(target, anchors, ws, out);

    dim3 grid((CELLS_B + 255) / 256, NB);
    yolo_main<<<grid, 256, 0, stream>>>(output, ws, anchors, out);
}